// WaveGateHSM_64321430225015
// MI455X (gfx1250) — compile-verified
//
#include <hip/hip_runtime.h>
#include <math.h>

// ---------------- model constants ----------------
#define L_DIM 4
#define C_DIM 384
#define K_DIM 3
#define V_DIM 50281
#define CI_DIM 64
#define B_DIM 2
#define T_DIM 1024
#define BT_DIM (B_DIM * T_DIM)      // 2048
#define C3_DIM (3 * C_DIM)          // 1152
#define NT 4                        // N-tiles (16 cols each) per wave -> 16x64 strip

typedef __attribute__((ext_vector_type(2))) float v2f;
typedef __attribute__((ext_vector_type(8))) float v8f;

__device__ __forceinline__ float sigm(float x) { return 1.0f / (1.0f + expf(-x)); }

// f32 WMMA: D(16x16) = A(16x4) * B(4x16) + C   -> v_wmma_f32_16x16x4_f32
__device__ __forceinline__ v8f wmma4(v2f a, v2f b, v8f c) {
  return __builtin_amdgcn_wmma_f32_16x16x4_f32(
      /*neg_a=*/false, a, /*neg_b=*/false, b,
      /*c_mod=*/(short)0, c, /*reuse_a=*/false, /*reuse_b=*/false);
}

// ---------------- weight repack: conv_w [L,C,C,K] -> [L,K,C,C] ----------------
__global__ void repack_convw_kernel(const float* __restrict__ cw,
                                    float* __restrict__ wt, int n) {
  int i = blockIdx.x * blockDim.x + threadIdx.x;
  if (i >= n) return;
  int ci = i % C_DIM;
  int rest = i / C_DIM;
  int o = rest % C_DIM;
  rest /= C_DIM;
  int k = rest % K_DIM;
  int l = rest / K_DIM;
  wt[i] = cw[(((size_t)l * C_DIM + o) * C_DIM + ci) * K_DIM + k];
}

// ---------------- embedding gather ----------------
__global__ void embed_kernel(const int* __restrict__ idx,
                             const float* __restrict__ embed,
                             float* __restrict__ X, int n) {
  int i = blockIdx.x * blockDim.x + threadIdx.x;
  if (i >= n) return;
  int c = i % C_DIM;
  int bt = i / C_DIM;
  X[i] = embed[(size_t)idx[bt] * C_DIM + c];
}

// ---------------- layer norm (one block of 384 threads per row) ----------------
__device__ __forceinline__ float block_reduce_384(float v, float* buf) {
  int c = threadIdx.x;
  buf[c] = v;
  __syncthreads();
  if (c < 128) buf[c] += buf[c + 128] + buf[c + 256];
  __syncthreads();
  for (int s = 64; s > 0; s >>= 1) {
    if (c < s) buf[c] += buf[c + s];
    __syncthreads();
  }
  float r = buf[0];
  __syncthreads();
  return r;
}

__global__ void layernorm_kernel(const float* __restrict__ X, float* __restrict__ XN,
                                 const float* __restrict__ g, const float* __restrict__ b) {
  __shared__ float buf[C_DIM];
  int row = blockIdx.x;
  int c = threadIdx.x;
  float v = X[(size_t)row * C_DIM + c];
  float mean = block_reduce_384(v, buf) * (1.0f / C_DIM);
  float d = v - mean;
  float var = block_reduce_384(d * d, buf) * (1.0f / C_DIM);
  XN[(size_t)row * C_DIM + c] = d * rsqrtf(var + 1e-5f) * g[c] + b[c];
}

// ---------------- generic f32 WMMA GEMM: D[M,N] = A[M,K] @ W[N,K]^T (+bias)(+res) ----------
// one wave per 16x64 output strip (4 accumulators sharing one A fetch per K-step).
// K multiple of 4; M multiple of 16. Out-of-range W rows are CLAMPED (not masked):
// garbage feeds only output columns the store guard discards.
__global__ void gemm16_kernel(const float* __restrict__ A, const float* __restrict__ W,
                              const float* __restrict__ bias, const float* res,
                              float* D, int M, int N, int K,
                              int stripsN, int totalWaves) {
  int wid = blockIdx.x * (blockDim.x >> 5) + (threadIdx.x >> 5);
  if (wid >= totalWaves) return;                 // wave-uniform
  int mt = wid / stripsN, st = wid - mt * stripsN;
  int m0 = mt << 4, n0 = st << 6;
  int lane = threadIdx.x & 31;
  int half = lane >> 4;
  int r = lane & 15;
  const float* Arow = A + (size_t)(m0 + r) * K + (half << 1);
  const float* Wrow[NT];
#pragma unroll
  for (int j = 0; j < NT; ++j) {
    int nj = n0 + (j << 4) + r;
    int nc = nj < N ? nj : (N - 1);              // clamp, branchless
    Wrow[j] = W + (size_t)nc * K + (half << 1);
  }
  v8f acc[NT] = {};
  for (int k0 = 0; k0 < K; k0 += 4) {
    v2f a = *(const v2f*)(Arow + k0);
#pragma unroll
    for (int j = 0; j < NT; ++j) {
      v2f b = *(const v2f*)(Wrow[j] + k0);
      acc[j] = wmma4(a, b, acc[j]);
    }
  }
#pragma unroll
  for (int j = 0; j < NT; ++j) {
    int nj = n0 + (j << 4) + r;
    if (nj < N) {
      float badd = bias ? bias[nj] : 0.0f;
#pragma unroll
      for (int v = 0; v < 8; ++v) {
        int mm = m0 + v + (half << 3);
        size_t off = (size_t)mm * N + nj;
        float val = acc[j][v] + badd;
        if (res) val += res[off];
        D[off] = val;
      }
    }
  }
}

// ---------------- causal conv as 3 shifted WMMA GEMM passes, fused bias+residual ------------
// X[m,o] += sum_k sum_i WT[k][o][i] * XN[t-(2-k)][i] + bias[o]
// Causal-padding rows (t-shift < 0) are clamped to row 0 and zeroed by a hoisted 0/1 mask
// (no exec-mask manipulation in the inner loop).
__global__ void conv_gemm_kernel(const float* __restrict__ XN, const float* __restrict__ WT,
                                 const float* __restrict__ bias, float* X,
                                 int totalWaves) {
  const int N = C_DIM, K = C_DIM;
  const int stripsN = C_DIM / 64;                // 6
  int wid = blockIdx.x * (blockDim.x >> 5) + (threadIdx.x >> 5);
  if (wid >= totalWaves) return;
  int mt = wid / stripsN, st = wid - mt * stripsN;
  int m0 = mt << 4, n0 = st << 6;
  int lane = threadIdx.x & 31;
  int half = lane >> 4, r = lane & 15;
  int m = m0 + r;
  int bidx = m >> 10;                            // m / T_DIM
  int t = m & (T_DIM - 1);
  v8f acc[NT] = {};
  for (int kc = 0; kc < K_DIM; ++kc) {
    int shift = (K_DIM - 1) - kc;
    int ts = t - shift;
    float amask = ts >= 0 ? 1.0f : 0.0f;
    int tsc = ts >= 0 ? ts : 0;
    const float* Arow = XN + (size_t)(bidx * T_DIM + tsc) * K + (half << 1);
    const float* Wbase = WT + (size_t)kc * C_DIM * C_DIM + (half << 1);
    const float* Wrow[NT];
#pragma unroll
    for (int j = 0; j < NT; ++j) Wrow[j] = Wbase + (size_t)(n0 + (j << 4) + r) * K;
    for (int k0 = 0; k0 < K; k0 += 4) {
      v2f a = *(const v2f*)(Arow + k0);
      a.x *= amask;
      a.y *= amask;
#pragma unroll
      for (int j = 0; j < NT; ++j) {
        v2f b = *(const v2f*)(Wrow[j] + k0);
        acc[j] = wmma4(a, b, acc[j]);
      }
    }
  }
#pragma unroll
  for (int j = 0; j < NT; ++j) {
    int nj = n0 + (j << 4) + r;
    float badd = bias[nj];
#pragma unroll
    for (int v = 0; v < 8; ++v) {
      int mm = m0 + v + (half << 3);
      size_t off = (size_t)mm * N + nj;
      X[off] = X[off] + acc[j][v] + badd;
    }
  }
}

// ---------------- hierarchical state memory scan (sequential over T, one block per batch) ----
__global__ void hsm_scan_kernel(const float* __restrict__ P0, const float* __restrict__ W1,
                                const float* __restrict__ W2, float* __restrict__ HIST) {
  __shared__ float s0[C_DIM];
  __shared__ float s1[C_DIM];
  int b = blockIdx.x;
  int c = threadIdx.x;           // 384 threads, one channel each
  float my_s0 = 0.0f, my_s1 = 0.0f, my_s2 = 0.0f;
  s1[c] = 0.0f;
  __syncthreads();
  for (int t = 0; t < T_DIM; ++t) {
    const float* p = P0 + ((size_t)b * T_DIM + t) * C3_DIM;
    float f = p[c], i = p[C_DIM + c], g = p[2 * C_DIM + c];
    my_s0 = sigm(f) * my_s0 + sigm(i) * tanhf(g);
    s0[c] = my_s0;
    __syncthreads();
    if (((t + 1) & (CI_DIM - 1)) == 0) {       // level-1 commit every 64 steps
      const float* wf = W1 + (size_t)c * C_DIM;
      const float* wi = W1 + (size_t)(C_DIM + c) * C_DIM;
      const float* wg = W1 + (size_t)(2 * C_DIM + c) * C_DIM;
      float pf = 0.0f, pi = 0.0f, pg = 0.0f;
      for (int j = 0; j < C_DIM; ++j) {
        float s = s0[j];
        pf += wf[j] * s; pi += wi[j] * s; pg += wg[j] * s;
      }
      my_s1 = sigm(pf) * my_s1 + sigm(pi) * tanhf(pg);
      s1[c] = my_s1;
      __syncthreads();
    }
    if (((t + 1) & (CI_DIM * CI_DIM - 1)) == 0) {  // level-2 commit (never fires for T=1024)
      const float* wf = W2 + (size_t)c * C_DIM;
      const float* wi = W2 + (size_t)(C_DIM + c) * C_DIM;
      const float* wg = W2 + (size_t)(2 * C_DIM + c) * C_DIM;
      float pf = 0.0f, pi = 0.0f, pg = 0.0f;
      for (int j = 0; j < C_DIM; ++j) {
        float s = s1[j];
        pf += wf[j] * s; pi += wi[j] * s; pg += wg[j] * s;
      }
      my_s2 = sigm(pf) * my_s2 + sigm(pi) * tanhf(pg);
      __syncthreads();
    }
    float* h = HIST + ((size_t)b * T_DIM + t) * C3_DIM;
    h[c] = my_s0;
    h[C_DIM + c] = my_s1;
    h[2 * C_DIM + c] = my_s2;
  }
}

// ---------------- x += alpha * x * exp(-x^2/2) ----------------
__global__ void exptrans_kernel(float* __restrict__ X, const float* __restrict__ alpha,
                                int l, int n) {
  int i = blockIdx.x * blockDim.x + threadIdx.x;
  if (i >= n) return;
  float x = X[i];
  float a = alpha[l];
  X[i] = x + a * x * expf(-(x * x) * 0.5f);
}

// ---------------- launch ----------------
extern "C" void kernel_launch(void* const* d_in, const int* in_sizes, int n_in,
                              void* d_out, int out_size, void* d_ws, size_t ws_size,
                              hipStream_t stream) {
  (void)in_sizes; (void)n_in; (void)out_size; (void)ws_size;
  const int*   idx    = (const int*)d_in[0];
  const float* embed  = (const float*)d_in[1];
  const float* conv_w = (const float*)d_in[2];
  const float* conv_b = (const float*)d_in[3];
  const float* lvl_w  = (const float*)d_in[4];
  const float* out_w  = (const float*)d_in[5];
  const float* ln1_g  = (const float*)d_in[6];
  const float* ln1_b  = (const float*)d_in[7];
  const float* ln2_g  = (const float*)d_in[8];
  const float* ln2_b  = (const float*)d_in[9];
  const float* alpha  = (const float*)d_in[10];
  const float* head_w = (const float*)d_in[11];
  const float* head_b = (const float*)d_in[12];
  float* out = (float*)d_out;

  // workspace layout (floats)
  float* X    = (float*)d_ws;                         // [BT, C]      786432
  float* XN   = X    + (size_t)BT_DIM * C_DIM;        // [BT, C]      786432
  float* P0   = XN   + (size_t)BT_DIM * C_DIM;        // [BT, 3C]    2359296
  float* HIST = P0   + (size_t)BT_DIM * C3_DIM;       // [BT, 3C]    2359296
  float* WT   = HIST + (size_t)BT_DIM * C3_DIM;       // [L,K,C,C]   1769472

  const int nElemX = BT_DIM * C_DIM;

  // 0) repack conv weights [L,C,C,K] -> [L,K,C,C]
  {
    int n = L_DIM * K_DIM * C_DIM * C_DIM;
    repack_convw_kernel<<<(n + 255) / 256, 256, 0, stream>>>(conv_w, WT, n);
  }
  // 1) embedding gather
  embed_kernel<<<(nElemX + 255) / 256, 256, 0, stream>>>(idx, embed, X, nElemX);

  const int tilesM = BT_DIM / 16;                     // 128
  for (int l = 0; l < L_DIM; ++l) {
    // LN1
    layernorm_kernel<<<BT_DIM, C_DIM, 0, stream>>>(X, XN, ln1_g + l * C_DIM, ln1_b + l * C_DIM);
    // causal conv (3 shifted WMMA passes) + bias + residual into X
    {
      int strips = C_DIM / 64;                        // 6
      int total = tilesM * strips;                    // 768 waves
      conv_gemm_kernel<<<(total + 7) / 8, 256, 0, stream>>>(
          XN, WT + (size_t)l * K_DIM * C_DIM * C_DIM, conv_b + l * C_DIM, X, total);
    }
    // LN2
    layernorm_kernel<<<BT_DIM, C_DIM, 0, stream>>>(X, XN, ln2_g + l * C_DIM, ln2_b + l * C_DIM);
    // p0 = XN @ Wl0^T : [BT, 3C]
    {
      int strips = C3_DIM / 64;                       // 18
      int total = tilesM * strips;                    // 2304 waves
      gemm16_kernel<<<(total + 7) / 8, 256, 0, stream>>>(
          XN, lvl_w + (size_t)l * 3 * C3_DIM * C_DIM, nullptr, nullptr, P0,
          BT_DIM, C3_DIM, C_DIM, strips, total);
    }
    // sequential scan -> HIST
    hsm_scan_kernel<<<B_DIM, C_DIM, 0, stream>>>(
        P0,
        lvl_w + (size_t)l * 3 * C3_DIM * C_DIM + (size_t)1 * C3_DIM * C_DIM,
        lvl_w + (size_t)l * 3 * C3_DIM * C_DIM + (size_t)2 * C3_DIM * C_DIM,
        HIST);
    // X += HIST @ Wo^T  (Wo: [C, 3C])
    {
      int strips = C_DIM / 64;                        // 6
      int total = tilesM * strips;
      gemm16_kernel<<<(total + 7) / 8, 256, 0, stream>>>(
          HIST, out_w + (size_t)l * C_DIM * C3_DIM, nullptr, X, X,
          BT_DIM, C_DIM, C3_DIM, strips, total);
    }
    // exponential transform
    exptrans_kernel<<<(nElemX + 255) / 256, 256, 0, stream>>>(X, alpha, l, nElemX);
  }

  // head: out = X @ head_w^T + head_b  (N = 50281, last strip clamped + store-guarded)
  {
    int strips = (V_DIM + 63) / 64;                   // 786
    int total = tilesM * strips;                      // 100608 waves
    gemm16_kernel<<<(total + 7) / 8, 256, 0, stream>>>(
        X, head_w, head_b, nullptr, out, BT_DIM, V_DIM, C_DIM, strips, total);
  }
}